// NumericalEncodingLayer_65764539236820
// MI455X (gfx1250) — compile-verified
//
#include <hip/hip_runtime.h>
#include <stdint.h>

// out[bt, i, :] = relu(x[bt,i] * W[i,:] + b[i,:])
// B=8192, K=128, D=128  ->  512 MB f32 output stream; HBM-store bound (~22 us floor @ 23.3 TB/s)

typedef float        v4f __attribute__((ext_vector_type(4)));
typedef unsigned int v4u __attribute__((ext_vector_type(4)));
typedef int          v4i __attribute__((ext_vector_type(4)));
typedef int          v8i __attribute__((ext_vector_type(8)));

#define NB 8192
#define NK 128
#define ND 128

#if defined(__gfx1250__) && __has_builtin(__builtin_amdgcn_tensor_load_to_lds) && __has_builtin(__builtin_amdgcn_s_wait_tensorcnt)
#define USE_TDM 1
#else
#define USE_TDM 0
#endif

#if USE_TDM
// 1-D TDM copy of nelem f32 (nelem <= 65535) from global to LDS.
// D# layout per CDNA5 ISA:
//   group0: [1:0]=count=1, [63:32]=lds_addr, [120:64]=global_addr, [127:126]=type=2
//   group1: [17:16]=data_size=2 (4B), dim0=tile0=nelem, dim1=tile1=1, stride0=nelem
__device__ __forceinline__ void tdm_load_1d_f32(const float* gsrc, float* lds_dst, uint32_t nelem) {
  uint64_t ga  = (uint64_t)(uintptr_t)gsrc;
  uint32_t lds = (uint32_t)(uintptr_t)lds_dst;  // low 32 bits of generic LDS ptr = LDS byte addr

  v4u g0;
  g0.x = 1u;                                           // count = 1 (valid descriptor)
  g0.y = lds;                                          // lds_addr
  g0.z = (uint32_t)(ga & 0xffffffffu);                 // global_addr[31:0]
  g0.w = (uint32_t)((ga >> 32) & 0x1ffffffu)           // global_addr[56:32]
       | (2u << 30);                                   // type = 2 ("image")

  v8i g1;
  g1[0] = (int)(2u << 16);                             // wg_mask=0 | data_size=2 (4 bytes)
  g1[1] = (int)((nelem & 0xffffu) << 16);              // barrier_addr=0 | tensor_dim0[15:0]
  g1[2] = (int)(((nelem >> 16) & 0xffffu) | (1u << 16)); // tensor_dim0[31:16] | tensor_dim1[15:0]=1
  g1[3] = (int)((nelem & 0xffffu) << 16);              // tensor_dim1[31:16]=0 | tile_dim0=nelem
  g1[4] = 1;                                           // tile_dim1=1 | tile_dim2=0
  g1[5] = (int)nelem;                                  // tensor_dim0_stride[31:0]
  g1[6] = 0;                                           // stride0[47:32] | tensor_dim1_stride lo
  g1[7] = 0;                                           // tensor_dim1_stride hi

  v4i z4 = {0, 0, 0, 0};
#if defined(__clang_major__) && (__clang_major__ >= 23)
  v8i z8 = {0, 0, 0, 0, 0, 0, 0, 0};
  __builtin_amdgcn_tensor_load_to_lds(g0, g1, z4, z4, z8, 0);
#else
  __builtin_amdgcn_tensor_load_to_lds(g0, g1, z4, z4, 0);
#endif
}
#endif // USE_TDM

__global__ __launch_bounds__(256) void encode_relu_stream_kernel(
    const float* __restrict__ x,   // [NB*NK]
    const float* __restrict__ W,   // [NK*ND]
    const float* __restrict__ b,   // [NK*ND]
    float* __restrict__ out)       // [NB*NK*ND]
{
  __shared__ __align__(16) float sW[NK * ND];   // 64 KB
  __shared__ __align__(16) float sB[NK * ND];   // 64 KB  (128 KB total -> 2 blocks / WGP)

#if USE_TDM
  // TDM issues regardless of EXEC, so only wave 0 may reach these instructions
  // (wave-uniform branch -> other waves jump around them entirely).
  if (threadIdx.x < 32) {
    tdm_load_1d_f32(W, sW, NK * ND);
    tdm_load_1d_f32(b, sB, NK * ND);
    __builtin_amdgcn_s_wait_tensorcnt(0);       // TENSORcnt == 0: both DMAs landed in LDS
  }
  __syncthreads();
#else
  for (int i = threadIdx.x; i < NK * ND; i += 256) {
    sW[i] = W[i];
    sB[i] = b[i];
  }
  __syncthreads();
#endif

  // One float4 of the output per thread per iteration.
  // 32 consecutive lanes (one wave) cover exactly one 512 B output row -> one
  // coalesced global_store_b128 burst per lane, non-temporal (write-once stream).
  const uint32_t total4 = (uint32_t)NB * NK * (ND / 4);     // 33,554,432
  const uint32_t stride = gridDim.x * blockDim.x;
  for (uint32_t idx = blockIdx.x * blockDim.x + threadIdx.x; idx < total4; idx += stride) {
    const uint32_t row = idx >> 5;          // flat (bt*NK + i), uniform across the wave
    const uint32_t c4  = idx & 31u;         // float4 column within the row
    const float    xv  = x[row];            // broadcast scalar (one cacheline per wave)
    const uint32_t i   = row & (NK - 1);

    const v4f w  = *reinterpret_cast<const v4f*>(sW + i * ND + c4 * 4);  // ds_load_b128
    const v4f bb = *reinterpret_cast<const v4f*>(sB + i * ND + c4 * 4);

    v4f o;
    o.x = fmaxf(fmaf(xv, w.x, bb.x), 0.0f);
    o.y = fmaxf(fmaf(xv, w.y, bb.y), 0.0f);
    o.z = fmaxf(fmaf(xv, w.z, bb.z), 0.0f);
    o.w = fmaxf(fmaf(xv, w.w, bb.w), 0.0f);

    __builtin_nontemporal_store(o, reinterpret_cast<v4f*>(out) + idx);
  }
}

extern "C" void kernel_launch(void* const* d_in, const int* in_sizes, int n_in,
                              void* d_out, int out_size, void* d_ws, size_t ws_size,
                              hipStream_t stream) {
  (void)in_sizes; (void)n_in; (void)d_ws; (void)ws_size; (void)out_size;
  const float* x = (const float*)d_in[0];
  const float* W = (const float*)d_in[1];
  const float* b = (const float*)d_in[2];
  float* out     = (float*)d_out;

  // 1024 blocks x 256 threads: enough to saturate HBM stores while keeping
  // redundant per-block W/b LDS fills down to 128 MB of L2-resident reads.
  dim3 grid(1024), block(256);
  hipLaunchKernelGGL(encode_relu_stream_kernel, grid, block, 0, stream, x, W, b, out);
}